// ContextualCompressor_4234837754155
// MI455X (gfx1250) — compile-verified
//
#include <hip/hip_runtime.h>
#include <hip/hip_bf16.h>
#include <math.h>

#define BATCH  16
#define SEQ    512
#define DIM    2048
#define HDIM   1024
#define NPAIRS 128
#define NUN    256     // SEQ - 2*NPAIRS
#define OUTSEQ 384     // NPAIRS + NUN

#define KBLK    16
#define ASTRIDE 20     // A-tile LDS row stride (16 data + 4 pad dwords, 16B aligned, bank-spread)
#define BSTRIDE 72     // B-tile LDS row stride (64 data + 8 pad dwords => half-wave bank split)

typedef __attribute__((ext_vector_type(2))) float v2f;
typedef __attribute__((ext_vector_type(8))) float v8f;
typedef __attribute__((ext_vector_type(4))) unsigned int v4u;
typedef __attribute__((ext_vector_type(8))) int v8i_t;
typedef __attribute__((ext_vector_type(4))) int v4i_t;

#if defined(__has_builtin)
#if __has_builtin(__builtin_amdgcn_tensor_load_to_lds) && __has_builtin(__builtin_amdgcn_s_wait_tensorcnt)
#define USE_TDM 1
#endif
#endif

__device__ __forceinline__ float gelu_erf(float v) {
    return 0.5f * v * (1.0f + erff(v * 0.70710678118654752f));
}

__device__ __forceinline__ unsigned lds_off(const void* p) {
    // flat LDS aperture: addr[31:0] is the LDS byte offset
    return (unsigned)(size_t)p;
}

#ifdef USE_TDM
// Issue one TDM 2D tile load: tile1 rows of tile0 dwords, row stride strideElems
// dwords, into LDS at ldsOff with (pad_interval_enc, pad_amount_enc) LDS padding.
// This toolchain exposes the 6-arg builtin:
//   (uint32x4 g0, int32x8 g1, int32x4 g2, int32x4 g3, int32x8, i32 cpol)
__device__ __forceinline__ void tdm_tile_load(const float* g, unsigned ldsOff,
                                              unsigned tile0, unsigned tile1,
                                              unsigned strideElems,
                                              unsigned padIntEnc, unsigned padAmtEnc) {
    const unsigned long long ga = (unsigned long long)(size_t)g;
    v4u g0;
    g0[0] = 1u;                                        // count=1, user descriptor
    g0[1] = ldsOff;                                    // lds_addr
    g0[2] = (unsigned)(ga & 0xFFFFFFFFu);              // global_addr[31:0]
    g0[3] = (unsigned)((ga >> 32) & 0x1FFFFFFu) | (2u << 30); // addr[56:32] | type=2
    const unsigned td0 = 0x40000000u, td1 = 0x40000000u;      // huge => no OOB clip
    v8i_t g1;
    g1[0] = (int)((2u << 16) | (1u << 20) | (padIntEnc << 22) | (padAmtEnc << 25));
    g1[1] = (int)((td0 & 0xFFFFu) << 16);              // tensor_dim0[15:0]
    g1[2] = (int)(((td0 >> 16) & 0xFFFFu) | ((td1 & 0xFFFFu) << 16));
    g1[3] = (int)(((td1 >> 16) & 0xFFFFu) | (tile0 << 16));   // tile_dim0
    g1[4] = (int)(tile1 & 0xFFFFu);                    // tile_dim1 (tile_dim2=0)
    g1[5] = (int)strideElems;                          // tensor_dim0_stride[31:0]
    g1[6] = 0;
    g1[7] = 0;
    v4i_t z4 = {};
    v8i_t z8 = {};
    __builtin_amdgcn_tensor_load_to_lds(g0, g1, z4, z4, z8, 0);
}
#define TDM_WAIT(n) do { if (threadIdx.x < 32) __builtin_amdgcn_s_wait_tensorcnt(n); } while (0)
#else
#define TDM_WAIT(n) do { } while (0)
#endif

// ---- tile staging: 64 rows x 16 dwords (A-layout, stride ASTRIDE) -----------
__device__ __forceinline__ void stage_rows16(const float* g, int ld, float* dst) {
#ifdef USE_TDM
    if (threadIdx.x < 32) tdm_tile_load(g, lds_off(dst), 16, 64, (unsigned)ld, 3, 3);
#else
    const int t = threadIdx.x, r = t >> 2, c4 = (t & 3) * 4;
    *(float4*)&dst[r * ASTRIDE + c4] = *(const float4*)&g[(size_t)r * ld + c4];
#endif
}

// ---- tile staging: 16 rows x 64 dwords (B-layout, stride BSTRIDE) -----------
__device__ __forceinline__ void stage_cols64(const float* g, int ld, float* dst) {
#ifdef USE_TDM
    if (threadIdx.x < 32) tdm_tile_load(g, lds_off(dst), 64, 16, (unsigned)ld, 5, 7);
#else
    const int t = threadIdx.x, r = t >> 4, c4 = (t & 15) * 4;
    *(float4*)&dst[r * BSTRIDE + c4] = *(const float4*)&g[(size_t)r * ld + c4];
#endif
}

// ---- gathered-pair A staging for merger GEMM1 (always cooperative manual) ---
__device__ __forceinline__ void stage_pairA(const float* __restrict__ x,
                                            const int* __restrict__ pi,
                                            const int* __restrict__ pj,
                                            int m0blk, int k0, float* dst) {
    const int t = threadIdx.x, r = t >> 2, c4 = (t & 3) * 4;
    const int mrow = m0blk + r;
    const int b = mrow >> 7, p = mrow & 127;
    const int* src = (k0 < DIM) ? pi : pj;
    const int idx = src[b * NPAIRS + p];
    const int kk = (k0 < DIM) ? k0 : (k0 - DIM);
    *(float4*)&dst[r * ASTRIDE + c4] =
        *(const float4*)&x[((size_t)b * SEQ + idx) * DIM + kk + c4];
}

// ---- per-wave MAC over one K-block, B in [k][n] LDS layout ------------------
__device__ __forceinline__ void tile_mac_wB(const float* sA, const float* sB,
                                            v8f& acc0, v8f& acc1,
                                            int mloc, int n0, int n1, int mr, int kh) {
#pragma unroll
    for (int kk = 0; kk < KBLK; kk += 4) {
        const int ka = kk + kh * 2;
        v2f a, b0, b1;
        a.x = sA[(mloc + mr) * ASTRIDE + ka];
        a.y = sA[(mloc + mr) * ASTRIDE + ka + 1];
        b0.x = sB[ka * BSTRIDE + n0 + mr];
        b0.y = sB[(ka + 1) * BSTRIDE + n0 + mr];
        b1.x = sB[ka * BSTRIDE + n1 + mr];
        b1.y = sB[(ka + 1) * BSTRIDE + n1 + mr];
        acc0 = __builtin_amdgcn_wmma_f32_16x16x4_f32(false, a, false, b0, (short)0, acc0, false, false);
        acc1 = __builtin_amdgcn_wmma_f32_16x16x4_f32(false, a, false, b1, (short)0, acc1, false, false);
    }
}

// ---- per-wave MAC, B stored untransposed in A-layout (B = x^T) --------------
__device__ __forceinline__ void tile_mac_xB(const float* sA, const float* sBx,
                                            v8f& acc0, v8f& acc1,
                                            int mloc, int n0, int n1, int mr, int kh) {
#pragma unroll
    for (int kk = 0; kk < KBLK; kk += 4) {
        const int ka = kk + kh * 2;
        v2f a, b0, b1;
        a.x = sA[(mloc + mr) * ASTRIDE + ka];
        a.y = sA[(mloc + mr) * ASTRIDE + ka + 1];
        b0.x = sBx[(n0 + mr) * ASTRIDE + ka];
        b0.y = sBx[(n0 + mr) * ASTRIDE + ka + 1];
        b1.x = sBx[(n1 + mr) * ASTRIDE + ka];
        b1.y = sBx[(n1 + mr) * ASTRIDE + ka + 1];
        acc0 = __builtin_amdgcn_wmma_f32_16x16x4_f32(false, a, false, b0, (short)0, acc0, false, false);
        acc1 = __builtin_amdgcn_wmma_f32_16x16x4_f32(false, a, false, b1, (short)0, acc1, false, false);
    }
}

// ---------------- Stage 1: h1 = gelu(x @ imp_w1 + imp_b1)  [8192,1024] -------
__global__ void k_imp_gemm1(const float* __restrict__ x, const float* __restrict__ w1,
                            const float* __restrict__ b1, float* __restrict__ h1) {
    __shared__ float sA[2][64 * ASTRIDE];
    __shared__ float sB[2][16 * BSTRIDE];
    const int n0blk = blockIdx.x * 64, m0blk = blockIdx.y * 64;
    const int w = threadIdx.x >> 5, lane = threadIdx.x & 31;
    const int mr = lane & 15, kh = lane >> 4;
    const int mloc = (w & 3) * 16, nh = w >> 2;
    v8f acc0 = {}, acc1 = {};
    const int NKB = DIM / KBLK;
    stage_rows16(x + (size_t)m0blk * DIM, DIM, sA[0]);
    stage_cols64(w1 + n0blk, HDIM, sB[0]);
    int cur = 0;
    for (int kb = 0; kb < NKB - 1; ++kb) {
        const int kn = (kb + 1) * KBLK;
        stage_rows16(x + (size_t)m0blk * DIM + kn, DIM, sA[cur ^ 1]);
        stage_cols64(w1 + (size_t)kn * HDIM + n0blk, HDIM, sB[cur ^ 1]);
        TDM_WAIT(2);
        __syncthreads();
        tile_mac_wB(sA[cur], sB[cur], acc0, acc1, mloc, nh * 32, nh * 32 + 16, mr, kh);
        __syncthreads();
        cur ^= 1;
    }
    TDM_WAIT(0);
    __syncthreads();
    tile_mac_wB(sA[cur], sB[cur], acc0, acc1, mloc, nh * 32, nh * 32 + 16, mr, kh);
    const int c0 = n0blk + nh * 32 + mr, c1 = c0 + 16;
    const float bias0 = b1[c0], bias1 = b1[c1];
    for (int r = 0; r < 8; ++r) {
        const int row = m0blk + mloc + r + 8 * kh;
        h1[(size_t)row * HDIM + c0] = gelu_erf(acc0[r] + bias0);
        h1[(size_t)row * HDIM + c1] = gelu_erf(acc1[r] + bias1);
    }
}

// ---------------- Stage 2: imp_raw = sigmoid(h1 . w2 + b2) -------------------
__global__ void k_imp_reduce(const float* __restrict__ h1, const float* __restrict__ w2,
                             const float* __restrict__ b2, float* __restrict__ imp_raw) {
    const int row = blockIdx.x * 8 + (threadIdx.x >> 5);
    const int lane = threadIdx.x & 31;
    const float* hr = h1 + (size_t)row * HDIM;
    float s = 0.0f;
    for (int c = lane; c < HDIM; c += 32) s += hr[c] * w2[c];
    for (int o = 16; o > 0; o >>= 1) s += __shfl_xor(s, o, 32);
    if (lane == 0) imp_raw[row] = 1.0f / (1.0f + __expf(-(s + b2[0])));
}

// ---------------- Stage 3: inverse row norms ---------------------------------
__global__ void k_norm(const float* __restrict__ x, float* __restrict__ invnorm) {
    const int row = blockIdx.x * 8 + (threadIdx.x >> 5);
    const int lane = threadIdx.x & 31;
    const float* xr = x + (size_t)row * DIM;
    float s = 0.0f;
    for (int c = lane; c < DIM; c += 32) { float v = xr[c]; s += v * v; }
    for (int o = 16; o > 0; o >>= 1) s += __shfl_xor(s, o, 32);
    if (lane == 0) invnorm[row] = 1.0f / fmaxf(sqrtf(s), 1e-12f);
}

// ------- Stage 4: per-batch min-max normalize, coeff = invnorm / max(.1,imp) -
__global__ void k_coeff(const float* __restrict__ imp_raw, const float* __restrict__ invnorm,
                        float* __restrict__ coeff) {
    __shared__ float smn[SEQ];
    __shared__ float smx[SEQ];
    const int b = blockIdx.x, t = threadIdx.x;
    const float v = imp_raw[b * SEQ + t];
    smn[t] = v; smx[t] = v;
    __syncthreads();
    for (int s = 256; s > 0; s >>= 1) {
        if (t < s) {
            smn[t] = fminf(smn[t], smn[t + s]);
            smx[t] = fmaxf(smx[t], smx[t + s]);
        }
        __syncthreads();
    }
    const float mn = smn[0], mx = smx[0];
    const float nv = (mx > mn) ? (v - mn) / (mx - mn) : v;
    coeff[b * SEQ + t] = invnorm[b * SEQ + t] / fmaxf(nv, 0.1f);
}

// ------- Stage 5: score[b,i,j] = (x_i . x_j) * a_i * a_j, diag = -1 ----------
__global__ void k_score(const float* __restrict__ x, const float* __restrict__ coeff,
                        float* __restrict__ score) {
    __shared__ float sA[2][64 * ASTRIDE];
    __shared__ float sBx[2][64 * ASTRIDE];
    const int n0blk = blockIdx.x * 64, m0blk = blockIdx.y * 64, b = blockIdx.z;
    const int w = threadIdx.x >> 5, lane = threadIdx.x & 31;
    const int mr = lane & 15, kh = lane >> 4;
    const int mloc = (w & 3) * 16, nh = w >> 2;
    const float* xb = x + (size_t)b * SEQ * DIM;
    v8f acc0 = {}, acc1 = {};
    const int NKB = DIM / KBLK;
    stage_rows16(xb + (size_t)m0blk * DIM, DIM, sA[0]);
    stage_rows16(xb + (size_t)n0blk * DIM, DIM, sBx[0]);
    int cur = 0;
    for (int kb = 0; kb < NKB - 1; ++kb) {
        const int kn = (kb + 1) * KBLK;
        stage_rows16(xb + (size_t)m0blk * DIM + kn, DIM, sA[cur ^ 1]);
        stage_rows16(xb + (size_t)n0blk * DIM + kn, DIM, sBx[cur ^ 1]);
        TDM_WAIT(2);
        __syncthreads();
        tile_mac_xB(sA[cur], sBx[cur], acc0, acc1, mloc, nh * 32, nh * 32 + 16, mr, kh);
        __syncthreads();
        cur ^= 1;
    }
    TDM_WAIT(0);
    __syncthreads();
    tile_mac_xB(sA[cur], sBx[cur], acc0, acc1, mloc, nh * 32, nh * 32 + 16, mr, kh);
    const int c0 = n0blk + nh * 32 + mr, c1 = c0 + 16;
    const float aj0 = coeff[b * SEQ + c0], aj1 = coeff[b * SEQ + c1];
    float* sb = score + (size_t)b * SEQ * SEQ;
    for (int r = 0; r < 8; ++r) {
        const int row = m0blk + mloc + r + 8 * kh;
        const float ai = coeff[b * SEQ + row];
        float v0 = acc0[r] * ai * aj0;
        float v1 = acc1[r] * ai * aj1;
        if (row == c0) v0 = -1.0f;
        if (row == c1) v1 = -1.0f;
        sb[(size_t)row * SEQ + c0] = v0;
        sb[(size_t)row * SEQ + c1] = v1;
    }
}

// ------- Stage 6: greedy NMS-style pair selection (1 workgroup / batch) ------
__global__ void k_greedy(const float* __restrict__ score, int* __restrict__ pi,
                         int* __restrict__ pj, int* __restrict__ un) {
    __shared__ float rowmax[SEQ];
    __shared__ int   rowarg[SEQ];
    __shared__ int   avail[SEQ];
    __shared__ float rv[SEQ];
    __shared__ int   ri[SEQ];
    __shared__ int   sel[2];
    const int b = blockIdx.x;
    const int t = threadIdx.x;
    const float* sb = score + (size_t)b * SEQ * SEQ;
    avail[t] = 1;
    __syncthreads();
    {
        const float* sr = sb + (size_t)t * SEQ;
        float best = -1e30f; int barg = 0;
        for (int j = 0; j < SEQ; ++j) {
            const float v = sr[j];
            if (v > best) { best = v; barg = j; }
        }
        rowmax[t] = best; rowarg[t] = barg;
    }
    __syncthreads();
    for (int it = 0; it < NPAIRS; ++it) {
        rv[t] = avail[t] ? rowmax[t] : -1e30f;
        ri[t] = t;
        __syncthreads();
        for (int s = 256; s > 0; s >>= 1) {
            if (t < s) {
                if (rv[t + s] > rv[t]) { rv[t] = rv[t + s]; ri[t] = ri[t + s]; }
            }
            __syncthreads();
        }
        if (t == 0) {
            const int bi = ri[0];
            const int bj = rowarg[bi];
            sel[0] = bi; sel[1] = bj;
            avail[bi] = 0; avail[bj] = 0;
            pi[b * NPAIRS + it] = bi;
            pj[b * NPAIRS + it] = bj;
        }
        __syncthreads();
        const int bi = sel[0], bj = sel[1];
        if (avail[t] && (rowarg[t] == bi || rowarg[t] == bj)) {
            const float* sr = sb + (size_t)t * SEQ;
            float best = -1e30f; int barg = 0;
            for (int j = 0; j < SEQ; ++j) {
                if (!avail[j]) continue;
                const float v = sr[j];
                if (v > best) { best = v; barg = j; }
            }
            rowmax[t] = best; rowarg[t] = barg;
        }
        __syncthreads();
    }
    if (avail[t]) {
        int rank = 0;
        for (int j = 0; j < t; ++j) rank += avail[j];
        un[b * NUN + rank] = t;
    }
}

// ------- Stage 7: h2 = gelu(concat(x[pi],x[pj]) @ mrg_w1 + mrg_b1) -----------
__global__ void k_merge1(const float* __restrict__ x, const int* __restrict__ pi,
                         const int* __restrict__ pj, const float* __restrict__ w1,
                         const float* __restrict__ b1, float* __restrict__ h2) {
    __shared__ float sA[2][64 * ASTRIDE];
    __shared__ float sB[2][16 * BSTRIDE];
    const int n0blk = blockIdx.x * 64, m0blk = blockIdx.y * 64;
    const int w = threadIdx.x >> 5, lane = threadIdx.x & 31;
    const int mr = lane & 15, kh = lane >> 4;
    const int mloc = (w & 3) * 16, nh = w >> 2;
    v8f acc0 = {}, acc1 = {};
    const int NKB = (2 * DIM) / KBLK;
    stage_pairA(x, pi, pj, m0blk, 0, sA[0]);
    stage_cols64(w1 + n0blk, DIM, sB[0]);
    int cur = 0;
    for (int kb = 0; kb < NKB - 1; ++kb) {
        const int kn = (kb + 1) * KBLK;
        stage_pairA(x, pi, pj, m0blk, kn, sA[cur ^ 1]);
        stage_cols64(w1 + (size_t)kn * DIM + n0blk, DIM, sB[cur ^ 1]);
        TDM_WAIT(1);
        __syncthreads();
        tile_mac_wB(sA[cur], sB[cur], acc0, acc1, mloc, nh * 32, nh * 32 + 16, mr, kh);
        __syncthreads();
        cur ^= 1;
    }
    TDM_WAIT(0);
    __syncthreads();
    tile_mac_wB(sA[cur], sB[cur], acc0, acc1, mloc, nh * 32, nh * 32 + 16, mr, kh);
    const int c0 = n0blk + nh * 32 + mr, c1 = c0 + 16;
    const float bias0 = b1[c0], bias1 = b1[c1];
    for (int r = 0; r < 8; ++r) {
        const int row = m0blk + mloc + r + 8 * kh;
        h2[(size_t)row * DIM + c0] = gelu_erf(acc0[r] + bias0);
        h2[(size_t)row * DIM + c1] = gelu_erf(acc1[r] + bias1);
    }
}

// ------- Stage 8: merged = h2 @ mrg_w2 + mrg_b2 -> out[b, 0..127, :] ---------
__global__ void k_merge2(const float* __restrict__ h2, const float* __restrict__ w2,
                         const float* __restrict__ b2, float* __restrict__ out) {
    __shared__ float sA[2][64 * ASTRIDE];
    __shared__ float sB[2][16 * BSTRIDE];
    const int n0blk = blockIdx.x * 64, m0blk = blockIdx.y * 64;
    const int w = threadIdx.x >> 5, lane = threadIdx.x & 31;
    const int mr = lane & 15, kh = lane >> 4;
    const int mloc = (w & 3) * 16, nh = w >> 2;
    v8f acc0 = {}, acc1 = {};
    const int NKB = DIM / KBLK;
    stage_rows16(h2 + (size_t)m0blk * DIM, DIM, sA[0]);
    stage_cols64(w2 + n0blk, DIM, sB[0]);
    int cur = 0;
    for (int kb = 0; kb < NKB - 1; ++kb) {
        const int kn = (kb + 1) * KBLK;
        stage_rows16(h2 + (size_t)m0blk * DIM + kn, DIM, sA[cur ^ 1]);
        stage_cols64(w2 + (size_t)kn * DIM + n0blk, DIM, sB[cur ^ 1]);
        TDM_WAIT(2);
        __syncthreads();
        tile_mac_wB(sA[cur], sB[cur], acc0, acc1, mloc, nh * 32, nh * 32 + 16, mr, kh);
        __syncthreads();
        cur ^= 1;
    }
    TDM_WAIT(0);
    __syncthreads();
    tile_mac_wB(sA[cur], sB[cur], acc0, acc1, mloc, nh * 32, nh * 32 + 16, mr, kh);
    const int c0 = n0blk + nh * 32 + mr, c1 = c0 + 16;
    const float bias0 = b2[c0], bias1 = b2[c1];
    for (int r = 0; r < 8; ++r) {
        const int row = m0blk + mloc + r + 8 * kh;
        const int bb = row >> 7, p = row & 127;
        out[((size_t)bb * OUTSEQ + p) * DIM + c0] = acc0[r] + bias0;
        out[((size_t)bb * OUTSEQ + p) * DIM + c1] = acc1[r] + bias1;
    }
}

// ------- Stage 9: copy surviving tokens -> out[b, 128.., :] ------------------
__global__ void k_gather(const float* __restrict__ x, const int* __restrict__ un,
                         float* __restrict__ out) {
    const int u = blockIdx.x;
    const int b = blockIdx.y;
    const int src = un[b * NUN + u];
    const float4* s = (const float4*)(x + ((size_t)b * SEQ + src) * DIM);
    float4* d = (float4*)(out + ((size_t)b * OUTSEQ + NPAIRS + u) * DIM);
    for (int c = threadIdx.x; c < DIM / 4; c += blockDim.x) d[c] = s[c];
}

extern "C" void kernel_launch(void* const* d_in, const int* in_sizes, int n_in,
                              void* d_out, int out_size, void* d_ws, size_t ws_size,
                              hipStream_t stream) {
    const float* x      = (const float*)d_in[0];
    const float* imp_w1 = (const float*)d_in[1];
    const float* imp_b1 = (const float*)d_in[2];
    const float* imp_w2 = (const float*)d_in[3];
    const float* imp_b2 = (const float*)d_in[4];
    const float* mrg_w1 = (const float*)d_in[5];
    const float* mrg_b1 = (const float*)d_in[6];
    const float* mrg_w2 = (const float*)d_in[7];
    const float* mrg_b2 = (const float*)d_in[8];
    float* out = (float*)d_out;

    char* ws = (char*)d_ws;
    float* h1      = (float*)ws;                                    // 8192*1024 f
    float* h2      = h1;                                            // aliases h1
    float* imp_raw = (float*)(ws + (size_t)BATCH * SEQ * HDIM * 4); // 8192 f
    float* invnorm = imp_raw + BATCH * SEQ;
    float* coeff   = invnorm + BATCH * SEQ;
    float* score   = coeff + BATCH * SEQ;                           // 16*512*512 f
    int*   pi      = (int*)(score + (size_t)BATCH * SEQ * SEQ);
    int*   pj      = pi + BATCH * NPAIRS;
    int*   un      = pj + BATCH * NPAIRS;

    const dim3 blk256(256), blk512(512), blk128(128);

    k_imp_gemm1<<<dim3(HDIM / 64, (BATCH * SEQ) / 64), blk256, 0, stream>>>(x, imp_w1, imp_b1, h1);
    k_imp_reduce<<<dim3((BATCH * SEQ) / 8), blk256, 0, stream>>>(h1, imp_w2, imp_b2, imp_raw);
    k_norm<<<dim3((BATCH * SEQ) / 8), blk256, 0, stream>>>(x, invnorm);
    k_coeff<<<dim3(BATCH), blk512, 0, stream>>>(imp_raw, invnorm, coeff);
    k_score<<<dim3(SEQ / 64, SEQ / 64, BATCH), blk256, 0, stream>>>(x, coeff, score);
    k_greedy<<<dim3(BATCH), blk512, 0, stream>>>(score, pi, pj, un);
    k_merge1<<<dim3(DIM / 64, (BATCH * NPAIRS) / 64), blk256, 0, stream>>>(x, pi, pj, mrg_w1, mrg_b1, h2);
    k_merge2<<<dim3(DIM / 64, (BATCH * NPAIRS) / 64), blk256, 0, stream>>>(h2, mrg_w2, mrg_b2, out);
    k_gather<<<dim3(NUN, BATCH), blk128, 0, stream>>>(x, un, out);
}